// SSIM_9990093931155
// MI455X (gfx1250) — compile-verified
//
#include <hip/hip_runtime.h>

// ---------------------------------------------------------------------------
// Fused SSIM loss for (8, 21, 512, 512) fp32, 11x11 Gaussian window (sigma=1.5),
// valid padding -> 502x502, loss = sum_c (1 - mean_{n,h,w} ssim_map[:, c]).
//
// Both separable blur passes are mapped onto V_WMMA_F32_16X16X4_F32 using
// banded weight matrices; one wave produces one 16x16 output tile for all
// five blurred quantities (p, t, p^2, t^2, p*t) and reduces the masked SSIM
// sum. Deterministic two-kernel reduction (no float atomics).
// ---------------------------------------------------------------------------

typedef __attribute__((ext_vector_type(2))) float v2f;
typedef __attribute__((ext_vector_type(8))) float v8f;

// Only check the device-side builtin during device compilation; the host pass
// (x86) never declares amdgcn builtins and must not trip this guard.
#if defined(__HIP_DEVICE_COMPILE__) && !__has_builtin(__builtin_amdgcn_wmma_f32_16x16x4_f32)
#error "__builtin_amdgcn_wmma_f32_16x16x4_f32 not available on this toolchain"
#endif

#define WMMA_F32(A, B, C) \
    __builtin_amdgcn_wmma_f32_16x16x4_f32(false, (A), false, (B), (short)0, (C), false, false)

// Normalized Gaussian kernel, sigma = 1.5, size 11 (precomputed in double).
__device__ __forceinline__ float bandv(int d) {
    constexpr float KERN[11] = {
        0.001028381f, 0.007598758f, 0.036000773f, 0.109360754f, 0.213005530f,
        0.266011716f,
        0.213005530f, 0.109360754f, 0.036000773f, 0.007598758f, 0.001028381f};
    float r = 0.0f;
#pragma unroll
    for (int i = 0; i < 11; ++i) r = (d == i) ? KERN[i] : r;
    return r;
}

// Layout constants
#define PSTRIDE 30    // transposed patch: [col][row], 28 rows padded to 30 (even -> b64 aligned)
#define VSTRIDE 34    // V staging: [row][col], 32 cols padded to 34 (even -> b64 aligned)
#define WAVE_LDS 2720 // floats per wave: max(2*32*PSTRIDE=1920, 5*16*VSTRIDE=2720)

__global__ __launch_bounds__(128) void ssim_tiles(const float* __restrict__ pred,
                                                  const float* __restrict__ targ,
                                                  float* __restrict__ partial) {
    __shared__ float smem[4][WAVE_LDS];

    const int lane = threadIdx.x & 31;
    const int wib = threadIdx.x >> 5;
    const unsigned wgid = blockIdx.x * 4u + wib; // 0 .. 172031
    const int plane = wgid >> 10;                // n*21 + c, 0..167
    const int tile = wgid & 1023;                // 0..1023
    const int tr = tile >> 5;
    const int tc = tile & 31;
    const int r0 = tr * 16;
    const int c0 = tc * 16;
    const int pbase = plane * (512 * 512);

    float* wls = smem[wib];
    float* patchP = wls;          // 32 cols x PSTRIDE (transposed)
    float* patchT = wls + 32 * PSTRIDE;

    // ------------------------------------------------------------------
    // Load 28x32 input patches (rows r0..r0+27 clamped, cols c0..c0+31).
    // Stored transposed: patch[col*PSTRIDE + row].
    // Branch is uniform per wave (tc==31 is the only clamped-column case),
    // so EXEC is all-ones at every WMMA below.
    // ------------------------------------------------------------------
    if (tc < 31) {
#pragma unroll
        for (int it = 0; it < 7; ++it) {
            int i = lane + it * 32;       // 0..223
            int row = i >> 3;             // 0..27
            int col = (i & 7) * 4;        // 0,4,..,28
            int gr = min(r0 + row, 511);
            int base = pbase + gr * 512 + (c0 + col);
            float4 pv = *(const float4*)(pred + base);
            float4 tv = *(const float4*)(targ + base);
            patchP[(col + 0) * PSTRIDE + row] = pv.x;
            patchP[(col + 1) * PSTRIDE + row] = pv.y;
            patchP[(col + 2) * PSTRIDE + row] = pv.z;
            patchP[(col + 3) * PSTRIDE + row] = pv.w;
            patchT[(col + 0) * PSTRIDE + row] = tv.x;
            patchT[(col + 1) * PSTRIDE + row] = tv.y;
            patchT[(col + 2) * PSTRIDE + row] = tv.z;
            patchT[(col + 3) * PSTRIDE + row] = tv.w;
        }
    } else {
#pragma unroll
        for (int it = 0; it < 7; ++it) {
            int i = lane + it * 32;
            int row = i >> 3;
            int col = (i & 7) * 4;
            int gr = min(r0 + row, 511);
#pragma unroll
            for (int u = 0; u < 4; ++u) {
                int gc = min(c0 + col + u, 511);
                int base = pbase + gr * 512 + gc;
                patchP[(col + u) * PSTRIDE + row] = pred[base];
                patchT[(col + u) * PSTRIDE + row] = targ[base];
            }
        }
    }

    // ------------------------------------------------------------------
    // Banded Gaussian weights; identical lane values serve as the A-operand
    // for the vertical pass and the B-operand for the horizontal pass.
    // w[j] lane value: band(lane&15, 4j + 2*(lane>>4) + {0,1}).
    // ------------------------------------------------------------------
    const int half = lane >> 4;
    const int mn = lane & 15;
    v2f w[7];
#pragma unroll
    for (int j = 0; j < 7; ++j) {
        int kb = 4 * j + 2 * half;
        w[j].x = bandv(kb - mn);
        w[j].y = bandv(kb + 1 - mn);
    }

    // ------------------------------------------------------------------
    // Vertical pass: V[q](16x32) = Wv(16x28) x Xq(28x32), K in chunks of 4.
    // Two 16x16 column tiles (ct); 5 quantities share each patch load.
    // ------------------------------------------------------------------
    v8f acc[5][2] = {};
#pragma unroll
    for (int j = 0; j < 7; ++j) {
        int k0 = 4 * j + 2 * half;
#pragma unroll
        for (int ct = 0; ct < 2; ++ct) {
            int coff = (16 * ct + mn) * PSTRIDE + k0;
            v2f pv = *(const v2f*)(patchP + coff);
            v2f tv = *(const v2f*)(patchT + coff);
            acc[0][ct] = WMMA_F32(w[j], pv, acc[0][ct]);
            acc[1][ct] = WMMA_F32(w[j], tv, acc[1][ct]);
            acc[2][ct] = WMMA_F32(w[j], pv * pv, acc[2][ct]);
            acc[3][ct] = WMMA_F32(w[j], tv * tv, acc[3][ct]);
            acc[4][ct] = WMMA_F32(w[j], pv * tv, acc[4][ct]);
        }
    }

    // Patches are dead now; reuse the wave's LDS region for V staging.
    asm volatile("s_wait_dscnt 0" ::: "memory");

    // Store V (D-layout -> row-major [m][col], stride VSTRIDE).
#pragma unroll
    for (int q = 0; q < 5; ++q) {
        float* Vq = wls + q * (16 * VSTRIDE);
#pragma unroll
        for (int ct = 0; ct < 2; ++ct) {
#pragma unroll
            for (int v = 0; v < 8; ++v) {
                Vq[(v + 8 * half) * VSTRIDE + 16 * ct + mn] = acc[q][ct][v];
            }
        }
    }

    asm volatile("s_wait_dscnt 0" ::: "memory");

    // ------------------------------------------------------------------
    // Horizontal pass: O[q](16x16) = V[q](16x28) x Wh(28x16).
    // A-operand pairs (k, k+1) are contiguous b64 loads from V staging.
    // ------------------------------------------------------------------
    v8f hres[5];
#pragma unroll
    for (int q = 0; q < 5; ++q) {
        const float* Vq = wls + q * (16 * VSTRIDE);
        v8f h = {};
#pragma unroll
        for (int j = 0; j < 7; ++j) {
            v2f a = *(const v2f*)(Vq + mn * VSTRIDE + 4 * j + 2 * half);
            h = WMMA_F32(a, w[j], h);
        }
        hres[q] = h;
    }

    // ------------------------------------------------------------------
    // SSIM map + masked sum over valid (row,col) < 502.
    // D layout: element (M = v + 8*half, N = lane&15).
    // ------------------------------------------------------------------
    const float c1 = 0.0001f; // (0.01*1.0)^2
    const float c2 = 0.0009f; // (0.03*1.0)^2
    float s = 0.0f;
#pragma unroll
    for (int v = 0; v < 8; ++v) {
        float mx = hres[0][v];
        float my = hres[1][v];
        float mxx = mx * mx, myy = my * my, mxy = mx * my;
        float sxx = hres[2][v] - mxx;
        float syy = hres[3][v] - myy;
        float sxy = hres[4][v] - mxy;
        float cs = (2.0f * sxy + c2) / (sxx + syy + c2);
        float ss = (2.0f * mxy + c1) / (mxx + myy + c1) * cs;
        int row = r0 + v + 8 * half;
        int col = c0 + mn;
        s += (row < 502 && col < 502) ? ss : 0.0f;
    }

    // Wave reduction (wave32) and deterministic partial write.
#pragma unroll
    for (int off = 16; off > 0; off >>= 1) s += __shfl_xor(s, off, 32);
    if (lane == 0) {
        int cls = plane % 21;
        int n = plane / 21;
        partial[cls * 8192 + n * 1024 + tile] = s;
    }
}

// Deterministic fixed-order final reduction: 21 classes x 8192 partials.
__global__ __launch_bounds__(256) void ssim_reduce(const float* __restrict__ partial,
                                                   float* __restrict__ out) {
    __shared__ float red[256];
    const int tid = threadIdx.x;
    const float inv_cnt = 1.0f / 2016032.0f; // 8 * 502 * 502
    float loss = 0.0f;
    for (int c = 0; c < 21; ++c) {
        float sgm = 0.0f;
        for (int i = tid; i < 8192; i += 256) sgm += partial[c * 8192 + i];
        red[tid] = sgm;
        __syncthreads();
        for (int st = 128; st > 0; st >>= 1) {
            if (tid < st) red[tid] += red[tid + st];
            __syncthreads();
        }
        if (tid == 0) loss += 1.0f - red[0] * inv_cnt;
        __syncthreads();
    }
    if (tid == 0) out[0] = loss;
}

extern "C" void kernel_launch(void* const* d_in, const int* in_sizes, int n_in,
                              void* d_out, int out_size, void* d_ws, size_t ws_size,
                              hipStream_t stream) {
    const float* pred = (const float*)d_in[0];
    const float* targ = (const float*)d_in[1];
    float* partial = (float*)d_ws; // 21 * 8192 floats = 688,128 B, fully rewritten per call

    // 168 planes * 1024 tiles = 172,032 waves; 4 waves per 128-thread block.
    ssim_tiles<<<43008, 128, 0, stream>>>(pred, targ, partial);
    ssim_reduce<<<1, 256, 0, stream>>>(partial, (float*)d_out);
}